// RoIPooling3D_63178968924300
// MI455X (gfx1250) — compile-verified
//
#include <hip/hip_runtime.h>

// Problem constants (from the reference setup): B=2, N=32, C=128, P=8192, G^3=8.
#define B_   2
#define N_   32
#define C_   128
#define P_   8192
#define G3_  8
#define NEGV (-1.0e30f)

// ---- CDNA5 async global->LDS path (guarded; falls back to plain loads) ----
#if defined(__AMDGCN__) && \
    __has_builtin(__builtin_amdgcn_global_load_async_to_lds_b32) && \
    __has_builtin(__builtin_amdgcn_s_wait_asynccnt)
#define USE_ASYNC_LDS 1
#else
#define USE_ASYNC_LDS 0
#endif

// Kernel A: per-box mask + rank-ordered compaction of point indices.
// grid = B*N blocks, 256 threads; thread t owns contiguous points [t*32, t*32+32).
__global__ __launch_bounds__(256) void roi_mask_scan(
    const float* __restrict__ boxes,   // (B,N,6) cx,cy,cz,sx,sy,sz
    const float* __restrict__ pc,      // (B,P,3)
    const int*   __restrict__ offset,  // (B)
    int* __restrict__ list,            // (B*N, P) compacted point indices
    int* __restrict__ cnt_out)         // (B*N)
{
  const int bn  = blockIdx.x;          // b*N + n
  const int b   = bn / N_;
  const int tid = threadIdx.x;

  const float* bx = boxes + (size_t)bn * 6;
  const float hx = bx[3] * 0.5f, hy = bx[4] * 0.5f, hz = bx[5] * 0.5f;
  const float lox = bx[0] - hx, hix = bx[0] + hx;
  const float loy = bx[1] - hy, hiy = bx[1] + hy;
  const float loz = bx[2] - hz, hiz = bx[2] + hz;
  const int off = offset[b];

  const int CH = P_ / 256;             // 32 points per thread
  const int p0 = tid * CH;
  const float* pcb = pc + (size_t)b * P_ * 3;

  unsigned mask = 0u;
  int cnt = 0;
  #pragma unroll 4
  for (int j = 0; j < CH; ++j) {
    const int p = p0 + j;
    const float x = pcb[(size_t)p * 3 + 0];
    const float y = pcb[(size_t)p * 3 + 1];
    const float z = pcb[(size_t)p * 3 + 2];
    const bool in = (x >= lox) & (x <= hix) &
                    (y >= loy) & (y <= hiy) &
                    (z >= loz) & (z <= hiz) & (p < off);
    mask |= (in ? 1u : 0u) << j;
    cnt  += in ? 1 : 0;
  }

  // Workgroup exclusive scan over per-thread counts (Hillis-Steele in LDS).
  __shared__ int scan[256];
  scan[tid] = cnt;
  __syncthreads();
  for (int s = 1; s < 256; s <<= 1) {
    const int v = scan[tid];
    const int a = (tid >= s) ? scan[tid - s] : 0;
    __syncthreads();
    scan[tid] = v + a;
    __syncthreads();
  }
  int base = scan[tid] - cnt;          // exclusive prefix for this thread

  int* lw = list + (size_t)bn * P_;
  for (int j = 0; j < CH; ++j) {
    if (mask & (1u << j)) lw[base++] = p0 + j;   // ascending p => ranks match cumsum
  }
  if (tid == 0) cnt_out[bn] = scan[255];
}

// Kernel B: per-channel max over the 8 rank-bins.
// grid = B*N blocks, 128 threads (one per channel, 4 wave32s).
__global__ __launch_bounds__(128) void roi_maxpool(
    const float* __restrict__ feat,    // (B,C,P)
    const int*   __restrict__ list,    // (B*N, P)
    const int*   __restrict__ cnt_in,  // (B*N)
    float* __restrict__ out)           // (B,N,C,8)
{
  const int bn  = blockIdx.x;
  const int b   = bn / N_;
  const int c   = threadIdx.x;         // 0..127
  const int cnt = cnt_in[bn];          // uniform across block

  float* ob = out + ((size_t)bn * C_ + c) * G3_;
  if (cnt == 0) {                      // uniform branch: no barrier divergence
    #pragma unroll
    for (int k = 0; k < G3_; ++k) ob[k] = 0.0f;
    return;
  }

  // Stage the compacted index list for this box into LDS (<= 32 KB).
  __shared__ int sidx[P_];
  const int* lw = list + (size_t)bn * P_;
#if USE_ASYNC_LDS
  for (int i = c; i < cnt; i += C_) {
    // (global src, lds dst, imm offset, imm cpol) — generic int* params,
    // clang inserts the AS1/AS3 addrspacecasts.
    __builtin_amdgcn_global_load_async_to_lds_b32(
        const_cast<int*>(lw + i), sidx + i, 0, 0);
  }
  __builtin_amdgcn_s_wait_asynccnt(0);
  __syncthreads();
#else
  for (int i = c; i < cnt; i += C_) sidx[i] = lw[i];
  __syncthreads();
#endif

  const float* fr = feat + ((size_t)b * C_ + c) * P_;  // this channel's 32KB row
  __builtin_prefetch(fr, 0, 0);        // -> global_prefetch_b8 (row is L2-hot anyway)

  #pragma unroll
  for (int k = 0; k < G3_; ++k) {
    const int start = (k * cnt) >> 3;              // floor(k*cnt/8)
    const int end   = ((k + 1) * cnt + 7) >> 3;    // ceil((k+1)*cnt/8)
    float m = NEGV;                                // bins are non-empty when cnt>0
    for (int r = start; r < end; ++r) {
      m = fmaxf(m, fr[sidx[r]]);
    }
    ob[k] = m;
  }
}

extern "C" void kernel_launch(void* const* d_in, const int* in_sizes, int n_in,
                              void* d_out, int out_size, void* d_ws, size_t ws_size,
                              hipStream_t stream) {
  (void)in_sizes; (void)n_in; (void)out_size; (void)ws_size;
  const float* boxes = (const float*)d_in[0];  // (B,N,6)
  const float* feat  = (const float*)d_in[1];  // (B,C,P)
  const int*   off   = (const int*)  d_in[2];  // (B)
  const float* pc    = (const float*)d_in[3];  // (B,P,3)
  float*       out   = (float*)d_out;          // (B,N,C,2,2,2)

  // Workspace layout: index list (B*N*P ints = 2 MB) + counts (B*N ints).
  int* list = (int*)d_ws;
  int* cnt  = list + (size_t)B_ * N_ * P_;

  roi_mask_scan<<<B_ * N_, 256, 0, stream>>>(boxes, pc, off, list, cnt);
  roi_maxpool <<<B_ * N_, 128, 0, stream>>>(feat, list, cnt, out);
}